// HybridDiceCLDiceLoss_58866821759051
// MI455X (gfx1250) — compile-verified
//
#include <hip/hip_runtime.h>
#include <math.h>

#define V1    2097152   // 128^3
#define NVOX  4194304   // 2 * 128^3
#define DSZ   128
#define PLANE 16384     // 128*128

typedef unsigned int v4u __attribute__((ext_vector_type(4)));
typedef int          v8i __attribute__((ext_vector_type(8)));
typedef int          v4i __attribute__((ext_vector_type(4)));

// ---------------- wave32 reduction helper ----------------
__device__ __forceinline__ float wave_sum(float v) {
    #pragma unroll
    for (int off = 16; off > 0; off >>= 1)
        v += __shfl_down(v, off, 32);
    return v;
}

// ---------------- TDM: DMA a 3D tile (tdx x tdy x tdz) into LDS ----------------
// D# built per CDNA5 ISA ch.8. One wave issues; TENSORcnt fences completion.
__device__ __forceinline__ void tdm_load_tile(const float* base, int csx, int csy, int csz,
                                              unsigned lds_off, int tdx, int tdy, int tdz) {
    const unsigned long long gaddr = (unsigned long long)(const void*)
        (base + (size_t)csz * PLANE + (size_t)csy * DSZ + (size_t)csx);
    v4u g0;
    g0[0] = 1u;                                                       // count=1
    g0[1] = lds_off;                                                  // lds_addr
    g0[2] = (unsigned)gaddr;                                          // global_addr[31:0]
    g0[3] = (unsigned)((gaddr >> 32) & 0x1FFFFFFull) | 0x80000000u;   // addr[56:32] | type=2
    v8i g1;
    g1[0] = (int)(2u << 16);                        // data_size=4B
    g1[1] = (int)(128u << 16);                      // tensor_dim0[15:0]=128
    g1[2] = (int)(128u << 16);                      // tensor_dim1[15:0]=128
    g1[3] = (int)((unsigned)tdx << 16);             // tile_dim0
    g1[4] = (int)((unsigned)tdy | ((unsigned)tdz << 16));  // tile_dim1 | tile_dim2
    g1[5] = 128;                                    // tensor_dim0_stride
    g1[6] = (int)(16384u << 16);                    // tensor_dim1_stride[15:0]
    g1[7] = 0;
    v4i g2; g2[0] = 128; g2[1] = 0; g2[2] = PLANE * DSZ; g2[3] = 0;   // tensor_dim2 etc.
    v4i g3; g3[0] = 0; g3[1] = 0; g3[2] = 0; g3[3] = 0;
    v8i g4; g4[0] = 0; g4[1] = 0; g4[2] = 0; g4[3] = 0; g4[4] = 0; g4[5] = 0; g4[6] = 0; g4[7] = 0;
    if (threadIdx.x < 32) {
        __builtin_amdgcn_tensor_load_to_lds(g0, g1, g2, g3, g4, 0);
        __builtin_amdgcn_s_wait_tensorcnt(0);
    }
}

// ---------------- accumulator zeroing ----------------
__global__ void zero_acc_kernel(float* acc) {
    if (threadIdx.x < 32) acc[threadIdx.x] = 0.0f;
}

// ---------------- softmax / CE / Dice stats + fg volumes ----------------
// acc: [0]=ce_sum, [1..4]=inter(n0c1,n0c2,n1c1,n1c2), [5..8]=sum_p, [9..12]=sum_t,
//      [13]=sum(skel_p*tfg), [14]=sum(skel_p), [15]=sum(skel_l*pfg), [16]=sum(skel_l)
__global__ void head_kernel(const float* __restrict__ pred, const int* __restrict__ tgt,
                            float* __restrict__ fgp, float* __restrict__ fgt,
                            float* __restrict__ acc) {
    __shared__ float sacc[13];
    if (threadIdx.x < 13) sacc[threadIdx.x] = 0.0f;
    __syncthreads();

    const size_t i = (size_t)blockIdx.x * blockDim.x + threadIdx.x;  // grid == NVOX/256
    const int    n = (int)(i >> 21);
    const size_t s = i & (size_t)(V1 - 1);
    const size_t b = (size_t)n * 3 * V1 + s;

    const float a0 = pred[b];
    const float a1 = pred[b + (size_t)V1];
    const float a2 = pred[b + 2 * (size_t)V1];
    const float m  = fmaxf(a0, fmaxf(a1, a2));
    const float e0 = __expf(a0 - m), e1 = __expf(a1 - m), e2 = __expf(a2 - m);
    const float den = e0 + e1 + e2;
    const float inv = 1.0f / den;
    const float p1 = e1 * inv, p2 = e2 * inv;

    const int t = tgt[i];
    const float lt = (t == 0 ? a0 : (t == 1 ? a1 : a2)) - m - __logf(den);

    fgp[i] = fminf(fmaxf(p1 + p2, 0.0f), 1.0f);
    fgt[i] = (t > 0) ? 1.0f : 0.0f;

    const int o = n * 2;
    float ce = -lt;
    float i1 = (t == 1) ? p1 : 0.0f;
    float i2 = (t == 2) ? p2 : 0.0f;
    float t1 = (t == 1) ? 1.0f : 0.0f;
    float t2 = (t == 2) ? 1.0f : 0.0f;
    float sp1 = p1, sp2 = p2;

    ce = wave_sum(ce); i1 = wave_sum(i1); i2 = wave_sum(i2);
    sp1 = wave_sum(sp1); sp2 = wave_sum(sp2);
    t1 = wave_sum(t1); t2 = wave_sum(t2);

    if ((threadIdx.x & 31) == 0) {
        atomicAdd(&sacc[0], ce);
        atomicAdd(&sacc[1 + o], i1);  atomicAdd(&sacc[2 + o], i2);
        atomicAdd(&sacc[5 + o], sp1); atomicAdd(&sacc[6 + o], sp2);
        atomicAdd(&sacc[9 + o], t1);  atomicAdd(&sacc[10 + o], t2);
    }
    __syncthreads();
    if (threadIdx.x < 13) atomicAdd(&acc[threadIdx.x], sacc[threadIdx.x]);
}

// =======================================================================
// Output tile per block: 32 x 8 x 2 (x,y,z), 256 threads, grid (4,16,128)
// =======================================================================

// ---------------- initial skeleton: skel = relu(x - maxpool(erode(x))) ----------------
// halo 2: T0 = 36x12x6 (from max(o-2,0)), T1 = erode -> 34x10x4 (from max(o-1,0))
__global__ void init_skel_kernel(const float* __restrict__ x, float* __restrict__ skel) {
    __shared__ float T0[6][12][36];
    __shared__ float T1[4][10][34];

    const int tid = threadIdx.x;
    const int ox = blockIdx.x * 32, oy = blockIdx.y * 8;
    const int n  = blockIdx.z >> 6, oz = (blockIdx.z & 63) * 2;
    const int c0x = max(ox - 2, 0), c0y = max(oy - 2, 0), c0z = max(oz - 2, 0);
    const int c1x = max(ox - 1, 0), c1y = max(oy - 1, 0), c1z = max(oz - 1, 0);

    const float* base = x + (size_t)n * V1;
    tdm_load_tile(base, c0x, c0y, c0z,
                  (unsigned)(unsigned long long)(const void*)&T0[0][0][0], 36, 12, 6);
    __syncthreads();

    // stage 1: erode(x) over 34x10x4 region
    for (int idx = tid; idx < 34 * 10 * 4; idx += 256) {
        const int lx = idx % 34, ly = (idx / 34) % 10, lz = idx / 340;
        const int qx = c1x + lx, qy = c1y + ly, qz = c1z + lz;
        const int xlo = max(qx - 1, 0), xhi = min(qx + 1, 127);
        const int ylo = max(qy - 1, 0), yhi = min(qy + 1, 127);
        const int zlo = max(qz - 1, 0), zhi = min(qz + 1, 127);
        float v = 1e30f;
        for (int z = zlo; z <= zhi; ++z)
            for (int y = ylo; y <= yhi; ++y)
                for (int xx = xlo; xx <= xhi; ++xx)
                    v = fminf(v, T0[z - c0z][y - c0y][xx - c0x]);
        T1[lz][ly][lx] = v;
    }
    __syncthreads();

    // stage 2: skel = relu(x - maxpool(T1))
    const int tx = tid & 31, ty = tid >> 5;
    for (int tz = 0; tz < 2; ++tz) {
        const int gx = ox + tx, gy = oy + ty, gz = oz + tz;
        const int xlo = max(gx - 1, 0), xhi = min(gx + 1, 127);
        const int ylo = max(gy - 1, 0), yhi = min(gy + 1, 127);
        const int zlo = max(gz - 1, 0), zhi = min(gz + 1, 127);
        float mo = -1e30f;
        for (int z = zlo; z <= zhi; ++z)
            for (int y = ylo; y <= yhi; ++y)
                for (int xx = xlo; xx <= xhi; ++xx)
                    mo = fmaxf(mo, T1[z - c1z][y - c1y][xx - c1x]);
        const float xv = T0[gz - c0z][gy - c0y][gx - c0x];
        const size_t vi = (size_t)n * V1 + (size_t)gz * PLANE + (size_t)gy * DSZ + (size_t)gx;
        skel[vi] = fmaxf(xv - mo, 0.0f);
    }
}

// ---------------- one fused soft-skel iteration ----------------
// xp = erode(src); o = maxpool(erode(xp)); delta = relu(xp - o);
// skel = clip(skel + relu(delta*(1-skel)), 0, 1)
// halo 3: T0 = 38x14x8, stage1(erode) T1 = 36x12x6, stage2(erode) T2 = 34x10x4
// FINAL: also accumulate sum(skel*other) -> acc[slot], sum(skel) -> acc[slot+1]
template <bool FINAL>
__global__ void fused_iter_kernel(const float* __restrict__ src, float* __restrict__ xp,
                                  float* __restrict__ skel, const float* __restrict__ other,
                                  float* __restrict__ acc, int slot) {
    __shared__ float T0[8][14][38];
    __shared__ float T1[6][12][36];
    __shared__ float T2[4][10][34];
    __shared__ float sv0, sv1;

    const int tid = threadIdx.x;
    const int ox = blockIdx.x * 32, oy = blockIdx.y * 8;
    const int n  = blockIdx.z >> 6, oz = (blockIdx.z & 63) * 2;
    const int c0x = max(ox - 3, 0), c0y = max(oy - 3, 0), c0z = max(oz - 3, 0);
    const int c1x = max(ox - 2, 0), c1y = max(oy - 2, 0), c1z = max(oz - 2, 0);
    const int c2x = max(ox - 1, 0), c2y = max(oy - 1, 0), c2z = max(oz - 1, 0);

    if (FINAL && tid == 0) { sv0 = 0.0f; sv1 = 0.0f; }

    const float* base = src + (size_t)n * V1;
    tdm_load_tile(base, c0x, c0y, c0z,
                  (unsigned)(unsigned long long)(const void*)&T0[0][0][0], 38, 14, 8);
    __syncthreads();

    // stage 1: T1 = erode(src) = xp (region 36x12x6)
    for (int idx = tid; idx < 36 * 12 * 6; idx += 256) {
        const int lx = idx % 36, ly = (idx / 36) % 12, lz = idx / 432;
        const int qx = c1x + lx, qy = c1y + ly, qz = c1z + lz;
        const int xlo = max(qx - 1, 0), xhi = min(qx + 1, 127);
        const int ylo = max(qy - 1, 0), yhi = min(qy + 1, 127);
        const int zlo = max(qz - 1, 0), zhi = min(qz + 1, 127);
        float v = 1e30f;
        for (int z = zlo; z <= zhi; ++z)
            for (int y = ylo; y <= yhi; ++y)
                for (int xx = xlo; xx <= xhi; ++xx)
                    v = fminf(v, T0[z - c0z][y - c0y][xx - c0x]);
        T1[lz][ly][lx] = v;
    }
    __syncthreads();

    // stage 2: T2 = erode(xp) (region 34x10x4)
    for (int idx = tid; idx < 34 * 10 * 4; idx += 256) {
        const int lx = idx % 34, ly = (idx / 34) % 10, lz = idx / 340;
        const int qx = c2x + lx, qy = c2y + ly, qz = c2z + lz;
        const int xlo = max(qx - 1, 0), xhi = min(qx + 1, 127);
        const int ylo = max(qy - 1, 0), yhi = min(qy + 1, 127);
        const int zlo = max(qz - 1, 0), zhi = min(qz + 1, 127);
        float v = 1e30f;
        for (int z = zlo; z <= zhi; ++z)
            for (int y = ylo; y <= yhi; ++y)
                for (int xx = xlo; xx <= xhi; ++xx)
                    v = fminf(v, T1[z - c1z][y - c1y][xx - c1x]);
        T2[lz][ly][lx] = v;
    }
    __syncthreads();

    // stage 3: outputs
    float v0 = 0.0f, v1 = 0.0f;
    const int tx = tid & 31, ty = tid >> 5;
    for (int tz = 0; tz < 2; ++tz) {
        const int gx = ox + tx, gy = oy + ty, gz = oz + tz;
        const int xlo = max(gx - 1, 0), xhi = min(gx + 1, 127);
        const int ylo = max(gy - 1, 0), yhi = min(gy + 1, 127);
        const int zlo = max(gz - 1, 0), zhi = min(gz + 1, 127);
        float mo = -1e30f;
        for (int z = zlo; z <= zhi; ++z)
            for (int y = ylo; y <= yhi; ++y)
                for (int xx = xlo; xx <= xhi; ++xx)
                    mo = fmaxf(mo, T2[z - c2z][y - c2y][xx - c2x]);
        const float xpv = T1[gz - c1z][gy - c1y][gx - c1x];
        const size_t vi = (size_t)n * V1 + (size_t)gz * PLANE + (size_t)gy * DSZ + (size_t)gx;
        xp[vi] = xpv;
        float sk = skel[vi];
        const float delta = fmaxf(xpv - mo, 0.0f);
        sk = fminf(fmaxf(sk + fmaxf(delta * (1.0f - sk), 0.0f), 0.0f), 1.0f);
        skel[vi] = sk;
        if (FINAL) { v0 += sk * other[vi]; v1 += sk; }
    }

    if (FINAL) {
        __syncthreads();  // sv0/sv1 initialized
        v0 = wave_sum(v0); v1 = wave_sum(v1);
        if ((tid & 31) == 0) { atomicAdd(&sv0, v0); atomicAdd(&sv1, v1); }
        __syncthreads();
        if (tid == 0) { atomicAdd(&acc[slot], sv0); atomicAdd(&acc[slot + 1], sv1); }
    }
}

// ---------------- final scalar ----------------
__global__ void finalize_kernel(const float* __restrict__ acc, float* __restrict__ out) {
    const float SD = 1e-5f, SC = 1e-5f;
    const float ce = acc[0] / (float)NVOX;
    float dsum = 0.0f;
    #pragma unroll
    for (int k = 0; k < 4; ++k) {
        const float inter = acc[1 + k], sp = acc[5 + k], st = acc[9 + k];
        dsum += 1.0f - (2.0f * inter + SD) / (st + sp + SD);
    }
    const float dice_ce = dsum * 0.25f + ce;
    const float tprec = (acc[13] + SC) / (acc[14] + SC);
    const float trec  = (acc[15] + SC) / (acc[16] + SC);
    const float cl    = 1.0f - 2.0f * tprec * trec / (tprec + trec + SC);
    out[0] = 1.0f * dice_ce + 0.3f * cl;
}

// ---------------- host orchestration ----------------
extern "C" void kernel_launch(void* const* d_in, const int* in_sizes, int n_in,
                              void* d_out, int out_size, void* d_ws, size_t ws_size,
                              hipStream_t stream) {
    (void)in_sizes; (void)n_in; (void)out_size; (void)ws_size;
    const float* pred   = (const float*)d_in[0];
    const int*   target = (const int*)d_in[1];
    float* out = (float*)d_out;

    float* acc  = (float*)d_ws;
    float* vols = acc + 64;
    float* FGP  = vols;
    float* FGT  = vols + 1 * (size_t)NVOX;
    float* SKEL = vols + 2 * (size_t)NVOX;
    float* XPa  = vols + 3 * (size_t)NVOX;
    float* XPb  = vols + 4 * (size_t)NVOX;

    zero_acc_kernel<<<1, 32, 0, stream>>>(acc);
    head_kernel<<<NVOX / 256, 256, 0, stream>>>(pred, target, FGP, FGT, acc);

    const dim3 fg(4, 16, 128);  // 32x8x2 output tile per 256-thread (8-wave) block

    auto soft_skel = [&](const float* x, const float* other, int slot) {
        init_skel_kernel<<<fg, 256, 0, stream>>>(x, SKEL);
        const float* s = x;
        float* A = XPa; float* B = XPb;
        for (int it = 0; it < 15; ++it) {
            if (it < 14)
                fused_iter_kernel<false><<<fg, 256, 0, stream>>>(s, A, SKEL, nullptr, nullptr, 0);
            else
                fused_iter_kernel<true><<<fg, 256, 0, stream>>>(s, A, SKEL, other, acc, slot);
            s = A; float* tswap = A; A = B; B = tswap;
        }
    };

    soft_skel(FGP, FGT, 13);  // skel_p vs target_fg
    soft_skel(FGT, FGP, 15);  // skel_l vs pred_fg

    finalize_kernel<<<1, 1, 0, stream>>>(acc, out);
}